// MTE_GCN_67473936220992
// MI455X (gfx1250) — compile-verified
//
#include <hip/hip_runtime.h>
#include <hip/hip_bf16.h>
#include <math.h>

// ---- problem constants (match reference) ----
#define NB   32      // batch
#define CIN  64      // input channels
#define TT   256     // time
#define VV   25      // vertices
#define KK   3       // kernels
#define KM   192     // K*MID = OUT of q/k/pre convs
#define OUTC 192     // final out channels
#define POS  (TT*VV) // 6400 positions per (n, channel)
#define EPSV 1e-5f

typedef float v2f __attribute__((ext_vector_type(2)));
typedef float v8f __attribute__((ext_vector_type(8)));

// ---------------------------------------------------------------
// Kernel 1: xbar[n,c,v] = mean_t x[n,c,t,v]      (51200 outputs)
// ---------------------------------------------------------------
__global__ void k_mean(const float* __restrict__ x, float* __restrict__ xbar) {
    int idx = blockIdx.x * 256 + threadIdx.x;          // < 32*64*25
    int v = idx % VV;
    int c = (idx / VV) % CIN;
    int n = idx / (VV * CIN);
    const float* xp = x + ((size_t)(n * CIN + c) * TT) * VV + v;
    float s = 0.f;
    #pragma unroll 8
    for (int t = 0; t < TT; ++t) s += xp[(size_t)t * VV];
    xbar[idx] = s * (1.0f / (float)TT);
}

// ---------------------------------------------------------------
// Kernel 2: hq[n,o,v] = wq_w[o,:]·xbar[n,:,v] + wq_b[o]; same for hw
// ---------------------------------------------------------------
__global__ void k_qk(const float* __restrict__ xbar,
                     const float* __restrict__ wq_w, const float* __restrict__ wq_b,
                     const float* __restrict__ wk_w, const float* __restrict__ wk_b,
                     float* __restrict__ hq, float* __restrict__ hw) {
    int idx = blockIdx.x * 256 + threadIdx.x;          // < 2*32*192*25
    const int total = NB * KM * VV;
    bool isK = idx >= total;
    int r = isK ? idx - total : idx;                   // r = (n*192+o)*25+v
    int v = r % VV;
    int o = (r / VV) % KM;
    int n = r / (VV * KM);
    const float* W = isK ? wk_w : wq_w;
    const float* B = isK ? wk_b : wq_b;
    const float* xb = xbar + (size_t)n * CIN * VV + v;
    float s = B[o];
    #pragma unroll 8
    for (int c = 0; c < CIN; ++c) s += W[o * CIN + c] * xb[c * VV];
    (isK ? hw : hq)[r] = s;
}

// ---------------------------------------------------------------
// Kernel 3: A_intra[n,k,v,w] = softmax_v( hq·hw ) * intra_kernel[k]
//   one block per (n,k); softmax normalizes over rows v per column w
// ---------------------------------------------------------------
__global__ void k_intra(const float* __restrict__ hq, const float* __restrict__ hw,
                        const float* __restrict__ intra_k, float* __restrict__ A_intra) {
    __shared__ float S[VV * VV];
    __shared__ float cmax[VV];
    __shared__ float csum[VV];
    int n = blockIdx.x / KK, k = blockIdx.x % KK;
    int tid = threadIdx.x;
    const float* q = hq + ((size_t)n * KM + k * 64) * VV;
    const float* p = hw + ((size_t)n * KM + k * 64) * VV;
    if (tid < VV * VV) {
        int v = tid / VV, w = tid % VV;
        float s = 0.f;
        #pragma unroll 8
        for (int c = 0; c < 64; ++c) s += q[c * VV + v] * p[c * VV + w];
        S[tid] = s;
    }
    __syncthreads();
    if (tid < VV) {                 // per-column w = tid reduction over v
        float mx = -INFINITY;
        for (int v = 0; v < VV; ++v) mx = fmaxf(mx, S[v * VV + tid]);
        float sm = 0.f;
        for (int v = 0; v < VV; ++v) sm += expf(S[v * VV + tid] - mx);
        cmax[tid] = mx; csum[tid] = sm;
    }
    __syncthreads();
    if (tid < VV * VV) {
        int w = tid % VV;
        A_intra[(size_t)blockIdx.x * (VV * VV) + tid] =
            expf(S[tid] - cmax[w]) / csum[w] * intra_k[k];
    }
}

// ---------------------------------------------------------------
// Kernel 4: A_inter = tanh(hq-hw)*inter_k ; A_extra = A_inter+A_intra (2nd output)
//           A_full  = A_extra + A[k]       (workspace, feeds spatial matmul)
// ---------------------------------------------------------------
__global__ void k_build_A(const float* __restrict__ hq, const float* __restrict__ hw,
                          const float* __restrict__ Ak, const float* __restrict__ A_intra,
                          const float* __restrict__ inter_k,
                          float* __restrict__ Afull, float* __restrict__ Aextra) {
    size_t idx = (size_t)blockIdx.x * 256 + threadIdx.x;   // < 32*3*64*625
    int w = (int)(idx % VV);       size_t r = idx / VV;
    int v = (int)(r % VV);         r /= VV;
    int mid = (int)(r % 64);       r /= 64;
    int k = (int)(r % KK);         int n = (int)(r / KK);
    int o = k * 64 + mid;
    float q  = hq[((size_t)n * KM + o) * VV + v];
    float kw = hw[((size_t)n * KM + o) * VV + w];
    float ai = tanhf(q - kw) * inter_k[k];
    float ex = ai + A_intra[((size_t)(n * KK + k)) * (VV * VV) + v * VV + w];
    Aextra[idx] = ex;
    Afull[idx]  = ex + Ak[k * (VV * VV) + v * VV + w];
}

// ---------------------------------------------------------------
// Kernel 5 (WMMA f32): h[n,o,pos] = relu( (s_o*pre_w)[o,:]·x[n,:,pos] + bias'_o )
//   block = 4 waves; wave owns a 16-row weight slab (BN-folded, cached
//   in 32 VGPRs) and loops 16 position tiles of 16.
// ---------------------------------------------------------------
__global__ void k_pre_gemm(const float* __restrict__ x,
                           const float* __restrict__ pre_w, const float* __restrict__ pre_b,
                           const float* __restrict__ pre_g, const float* __restrict__ pre_bb,
                           const float* __restrict__ pre_m, const float* __restrict__ pre_v,
                           float* __restrict__ h) {
    const int lane = threadIdx.x & 31;
    const int wave = threadIdx.x >> 5;
    const int n    = blockIdx.z;
    const int m0   = (blockIdx.y * 4 + wave) * 16;     // output-channel slab
    const int half = lane >> 4;                         // 0 / 1
    const int lcol = lane & 15;

    // A-fragment rows: this lane supplies row (m0+lcol), Ks (4k+2*half, +1)
    const int arow = m0 + lcol;
    const float sA = pre_g[arow] * rsqrtf(pre_v[arow] + EPSV);
    v2f afrag[16];
    #pragma unroll
    for (int ks = 0; ks < 16; ++ks) {
        const float* wp = pre_w + arow * CIN + ks * 4 + half * 2;
        afrag[ks].x = wp[0] * sA;
        afrag[ks].y = wp[1] * sA;
    }
    // per-lane store-row folded biases (C rows: half*8 + r)
    float brow[8];
    #pragma unroll
    for (int r = 0; r < 8; ++r) {
        int row = m0 + half * 8 + r;
        float s = pre_g[row] * rsqrtf(pre_v[row] + EPSV);
        brow[r] = s * pre_b[row] + pre_bb[row] - pre_m[row] * s;
    }

    const float* xn = x + (size_t)n * CIN * POS;
    float*       hn = h + (size_t)n * KM  * POS;

    for (int i = 0; i < 16; ++i) {
        const int p0 = blockIdx.x * 256 + i * 16;
        v8f acc = {0.f, 0.f, 0.f, 0.f, 0.f, 0.f, 0.f, 0.f};
        #pragma unroll
        for (int ks = 0; ks < 16; ++ks) {
            const int c = ks * 4 + half * 2;
            v2f b;
            b.x = xn[(size_t)c * POS + p0 + lcol];
            b.y = xn[(size_t)(c + 1) * POS + p0 + lcol];
            acc = __builtin_amdgcn_wmma_f32_16x16x4_f32(
                      false, afrag[ks], false, b, (short)0, acc, false, false);
        }
        #pragma unroll
        for (int r = 0; r < 8; ++r) {
            int row = m0 + half * 8 + r;
            hn[(size_t)row * POS + p0 + lcol] = fmaxf(acc[r] + brow[r], 0.f);
        }
    }
}

// ---------------------------------------------------------------
// Kernel 6: in-place spatial matmul  h[n,o,t,:] <- h[n,o,t,:] @ A_full[n,o]
//   block = (t-chunk of 32, o, n); h rows + 25x25 A staged in LDS.
// ---------------------------------------------------------------
__global__ void k_apply_A(float* __restrict__ h, const float* __restrict__ Afull) {
    __shared__ float hb[32 * VV];   // 800
    __shared__ float Ab[VV * VV];   // 625
    int n = blockIdx.z, o = blockIdx.y, tc = blockIdx.x;
    float*       hp = h     + ((size_t)n * KM + o) * POS + tc * (32 * VV);
    const float* Ap = Afull + ((size_t)n * KM + o) * (VV * VV);
    int tid = threadIdx.x;
    for (int i = tid; i < 32 * VV; i += 256) hb[i] = hp[i];
    for (int i = tid; i < VV * VV; i += 256) Ab[i] = Ap[i];
    __syncthreads();
    float acc[4];
    int cnt = 0;
    for (int i = tid; i < 32 * VV; i += 256) {
        int t = i / VV, w = i % VV;
        float s = 0.f;
        #pragma unroll
        for (int v = 0; v < VV; ++v) s += hb[t * VV + v] * Ab[v * VV + w];
        acc[cnt++] = s;
    }
    // reads were satisfied from LDS -> safe to overwrite global now
    cnt = 0;
    for (int i = tid; i < 32 * VV; i += 256) hp[i] = acc[cnt++];
}

// ---------------------------------------------------------------
// Kernel 7 (WMMA f32): out = relu( bn(post_w·h) + bn(res_w·x) )
//   one fused accumulator: 48 K-steps over h (192) + 16 K-steps over x (64),
//   BN scales folded into A-fragments, biases combined at epilogue.
// ---------------------------------------------------------------
__global__ void k_final_gemm(const float* __restrict__ x, const float* __restrict__ h,
                             const float* __restrict__ post_w, const float* __restrict__ post_b,
                             const float* __restrict__ bn_g,  const float* __restrict__ bn_b,
                             const float* __restrict__ bn_m,  const float* __restrict__ bn_v,
                             const float* __restrict__ res_w, const float* __restrict__ res_b,
                             const float* __restrict__ res_g, const float* __restrict__ res_bb,
                             const float* __restrict__ res_m, const float* __restrict__ res_v,
                             float* __restrict__ out) {
    const int lane = threadIdx.x & 31;
    const int wave = threadIdx.x >> 5;
    const int n    = blockIdx.z;
    const int m0   = (blockIdx.y * 4 + wave) * 16;
    const int half = lane >> 4;
    const int lcol = lane & 15;
    const int p0   = blockIdx.x * 16;

    const int arow = m0 + lcol;
    const float sp = bn_g[arow]  * rsqrtf(bn_v[arow]  + EPSV);
    const float sr = res_g[arow] * rsqrtf(res_v[arow] + EPSV);

    const float* hn = h + (size_t)n * KM  * POS;
    const float* xn = x + (size_t)n * CIN * POS;

    v8f acc = {0.f, 0.f, 0.f, 0.f, 0.f, 0.f, 0.f, 0.f};
    #pragma unroll 4
    for (int ks = 0; ks < KM / 4; ++ks) {            // post GEMM, K = 192
        const int o = ks * 4 + half * 2;
        v2f a;
        const float* wp = post_w + arow * KM + o;
        a.x = wp[0] * sp;  a.y = wp[1] * sp;
        v2f b;
        b.x = hn[(size_t)o * POS + p0 + lcol];
        b.y = hn[(size_t)(o + 1) * POS + p0 + lcol];
        acc = __builtin_amdgcn_wmma_f32_16x16x4_f32(
                  false, a, false, b, (short)0, acc, false, false);
    }
    #pragma unroll 4
    for (int ks = 0; ks < CIN / 4; ++ks) {           // residual GEMM, K = 64
        const int c = ks * 4 + half * 2;
        v2f a;
        const float* wp = res_w + arow * CIN + c;
        a.x = wp[0] * sr;  a.y = wp[1] * sr;
        v2f b;
        b.x = xn[(size_t)c * POS + p0 + lcol];
        b.y = xn[(size_t)(c + 1) * POS + p0 + lcol];
        acc = __builtin_amdgcn_wmma_f32_16x16x4_f32(
                  false, a, false, b, (short)0, acc, false, false);
    }
    #pragma unroll
    for (int r = 0; r < 8; ++r) {
        int row = m0 + half * 8 + r;
        float s1 = bn_g[row]  * rsqrtf(bn_v[row]  + EPSV);
        float s2 = res_g[row] * rsqrtf(res_v[row] + EPSV);
        float bias = s1 * post_b[row] + bn_b[row]  - bn_m[row]  * s1
                   + s2 * res_b[row]  + res_bb[row] - res_m[row] * s2;
        out[((size_t)n * OUTC + row) * POS + p0 + lcol] = fmaxf(acc[r] + bias, 0.f);
    }
}

// ---------------------------------------------------------------
// Host-side orchestration
// ---------------------------------------------------------------
extern "C" void kernel_launch(void* const* d_in, const int* in_sizes, int n_in,
                              void* d_out, int out_size, void* d_ws, size_t ws_size,
                              hipStream_t stream) {
    const float* x       = (const float*)d_in[0];
    const float* A       = (const float*)d_in[1];
    const float* wq_w    = (const float*)d_in[2];
    const float* wq_b    = (const float*)d_in[3];
    const float* wk_w    = (const float*)d_in[4];
    const float* wk_b    = (const float*)d_in[5];
    const float* pre_w   = (const float*)d_in[6];
    const float* pre_b   = (const float*)d_in[7];
    const float* pre_g   = (const float*)d_in[8];
    const float* pre_bb  = (const float*)d_in[9];
    const float* pre_m   = (const float*)d_in[10];
    const float* pre_v   = (const float*)d_in[11];
    const float* post_w  = (const float*)d_in[12];
    const float* post_b  = (const float*)d_in[13];
    const float* bn_g    = (const float*)d_in[14];
    const float* bn_b    = (const float*)d_in[15];
    const float* bn_m    = (const float*)d_in[16];
    const float* bn_v    = (const float*)d_in[17];
    const float* res_w   = (const float*)d_in[18];
    const float* res_b   = (const float*)d_in[19];
    const float* res_g   = (const float*)d_in[20];
    const float* res_bb  = (const float*)d_in[21];
    const float* res_m   = (const float*)d_in[22];
    const float* res_v   = (const float*)d_in[23];
    const float* intra_k = (const float*)d_in[24];
    const float* inter_k = (const float*)d_in[25];

    float* out_main  = (float*)d_out;                                    // 32*192*6400
    float* out_extra = (float*)d_out + (size_t)NB * OUTC * POS;          // 32*192*625

    // workspace layout (floats)
    float* ws       = (float*)d_ws;
    float* Afull    = ws;                                                //  3,840,000
    float* hbuf     = Afull + (size_t)NB * KM * VV * VV;                 // 39,321,600
    float* xbar     = hbuf  + (size_t)NB * KM * POS;                     //     51,200
    float* hq       = xbar  + (size_t)NB * CIN * VV;                     //    153,600
    float* hw       = hq    + (size_t)NB * KM * VV;                      //    153,600
    float* A_intra  = hw    + (size_t)NB * KM * VV;                      //     60,000

    // 1) time mean of x
    k_mean<<<dim3((NB * CIN * VV) / 256), dim3(256), 0, stream>>>(x, xbar);
    // 2) hq / hw
    k_qk<<<dim3((2 * NB * KM * VV) / 256), dim3(256), 0, stream>>>(
        xbar, wq_w, wq_b, wk_w, wk_b, hq, hw);
    // 3) intra attention + softmax
    k_intra<<<dim3(NB * KK), dim3(640), 0, stream>>>(hq, hw, intra_k, A_intra);
    // 4) tanh inter-term, A_extra output, A_full workspace
    k_build_A<<<dim3((NB * KM * VV * VV) / 256), dim3(256), 0, stream>>>(
        hq, hw, A, A_intra, inter_k, Afull, out_extra);
    // 5) h = relu(bn(pre conv))   [WMMA]
    k_pre_gemm<<<dim3(POS / 256, KM / 64, NB), dim3(128), 0, stream>>>(
        x, pre_w, pre_b, pre_g, pre_bb, pre_m, pre_v, hbuf);
    // 6) h <- h @ A_full (in place)
    k_apply_A<<<dim3(TT / 32, KM, NB), dim3(256), 0, stream>>>(hbuf, Afull);
    // 7) out = relu(bn(post conv) + bn(res conv))   [WMMA]
    k_final_gemm<<<dim3(POS / 16, OUTC / 64, NB), dim3(128), 0, stream>>>(
        x, hbuf, post_w, post_b, bn_g, bn_b, bn_m, bn_v,
        res_w, res_b, res_g, res_bb, res_m, res_v, out_main);
}